// Warp_10505490006605
// MI455X (gfx1250) — compile-verified
//
#include <hip/hip_runtime.h>

// 3D spatial-transformer warp (trilinear, border clamp), B=2, D=H=W=128, C=4.
// Memory-bound gather kernel: one thread per output voxel, float4 (b128)
// corner gathers, NT hints on the stream-once flo/out so L2 (192MB) caches x.

typedef float v4f __attribute__((ext_vector_type(4)));

__global__ __launch_bounds__(256) void warp3d_trilinear(
    const float* __restrict__ x,    // [B,128,128,128,4]
    const float* __restrict__ flo,  // [B,128,128,128,3]
    float* __restrict__ out,        // [B,128,128,128,4]
    int total)                      // B*D*H*W voxels
{
    int vox = blockIdx.x * blockDim.x + threadIdx.x;
    if (vox >= total) return;

    // Decompose voxel index (all dims 128, W fastest-varying, B slowest).
    int wi = vox & 127;
    int hi = (vox >> 7) & 127;
    int di = (vox >> 14) & 127;
    int bi = vox >> 21;

    // Flow: 3 floats per voxel, read exactly once -> non-temporal.
    const float* f = flo + (size_t)vox * 3;
    float fz = __builtin_nontemporal_load(f + 0);
    float fy = __builtin_nontemporal_load(f + 1);
    float fx = __builtin_nontemporal_load(f + 2);

    // Absolute sample coords, clamped to borders (matches jnp.clip(loc,0,max)).
    float z = fminf(fmaxf((float)di + fz, 0.0f), 127.0f);
    float y = fminf(fmaxf((float)hi + fy, 0.0f), 127.0f);
    float xc = fminf(fmaxf((float)wi + fx, 0.0f), 127.0f);

    float zf = floorf(z), yf = floorf(y), xf = floorf(xc);
    int z0 = (int)zf, y0 = (int)yf, x0 = (int)xf;
    int z1 = min(z0 + 1, 127);
    int y1 = min(y0 + 1, 127);
    int x1 = min(x0 + 1, 127);

    float wz1 = z - zf, wy1 = y - yf, wx1 = xc - xf;
    float wz0 = 1.0f - wz1, wy0 = 1.0f - wy1, wx0 = 1.0f - wx1;

    // Row bases in float4 units: ((b*128+z)*128+y)*128
    const v4f* __restrict__ xv = (const v4f*)x;
    int bbase = bi << 21;
    int r00 = bbase + (z0 << 14) + (y0 << 7);
    int r01 = bbase + (z0 << 14) + (y1 << 7);
    int r10 = bbase + (z1 << 14) + (y0 << 7);
    int r11 = bbase + (z1 << 14) + (y1 << 7);

    // 8 independent global_load_b128 gathers — issue all before consuming
    // so the compiler clusters them and uses partial loadcnt waits.
    v4f c000 = xv[r00 + x0];
    v4f c001 = xv[r00 + x1];
    v4f c010 = xv[r01 + x0];
    v4f c011 = xv[r01 + x1];
    v4f c100 = xv[r10 + x0];
    v4f c101 = xv[r10 + x1];
    v4f c110 = xv[r11 + x0];
    v4f c111 = xv[r11 + x1];

    float w00 = wz0 * wy0, w01 = wz0 * wy1;
    float w10 = wz1 * wy0, w11 = wz1 * wy1;
    float w000 = w00 * wx0, w001 = w00 * wx1;
    float w010 = w01 * wx0, w011 = w01 * wx1;
    float w100 = w10 * wx0, w101 = w10 * wx1;
    float w110 = w11 * wx0, w111 = w11 * wx1;

    v4f acc = c000 * w000;
    acc += c001 * w001;
    acc += c010 * w010;
    acc += c011 * w011;
    acc += c100 * w100;
    acc += c101 * w101;
    acc += c110 * w110;
    acc += c111 * w111;

    // Output written once -> non-temporal b128 store (keep L2 for x).
    __builtin_nontemporal_store(acc, (v4f*)out + vox);
}

extern "C" void kernel_launch(void* const* d_in, const int* in_sizes, int n_in,
                              void* d_out, int out_size, void* d_ws, size_t ws_size,
                              hipStream_t stream) {
    const float* x   = (const float*)d_in[0];   // B*D*H*W*4 floats
    const float* flo = (const float*)d_in[1];   // B*D*H*W*3 floats
    float* out       = (float*)d_out;

    int total = in_sizes[0] / 4;                // B*D*H*W voxels (C=4)
    int threads = 256;
    int blocks = (total + threads - 1) / threads;

    warp3d_trilinear<<<blocks, threads, 0, stream>>>(x, flo, out, total);
}